// TsMean_17051020165340
// MI455X (gfx1250) — compile-verified
//
#include <hip/hip_runtime.h>

// Sliding-window mean, window=20, stride=1, rows of 4096 f32.
// x: (128,256,4096) f32 -> out: (128,256,4077) f32. Pure bandwidth problem
// (~1 GB traffic -> ~44us floor @ 23.3 TB/s). Strategy: async-DMA each row
// into LDS (CDNA5 global_load_async_to_lds_b128, tracked by ASYNCcnt), then
// running-sum window reduction out of LDS, scalar coalesced stores.

#define ROW      4096
#define TOUT     4077   // (4096 - 20) / 1 + 1
#define WIN      20
#define NTHREADS 256
#define LDS_PAD  64     // cover tail window reads (max idx 4092+22=4114 < 4160)

__global__ __launch_bounds__(NTHREADS)
void TsMean_kernel(const float* __restrict__ x, float* __restrict__ out, int nrows) {
    __shared__ float row[ROW + LDS_PAD];

    const int r = blockIdx.x;
    if (r >= nrows) return;

    const float* __restrict__ src = x + (size_t)r * ROW;
    float* __restrict__ dst = out + (size_t)r * TOUT;
    const int t = (int)threadIdx.x;

    // Low 32 bits of a flat shared-memory pointer are the LDS byte offset
    // (flat->LDS mapping truncates to addr[31:0]).
    const unsigned lds_base = (unsigned)(uintptr_t)(void*)&row[0];

    // ---- Stage row into LDS via CDNA5 async loads (4 x b128 per thread) ----
    // Lane-contiguous 16B per lane per issue -> fully coalesced 16KB stream.
#pragma unroll
    for (int k = 0; k < 4; ++k) {
        unsigned ldso = lds_base + (unsigned)(t * 16 + k * 4096);
        const float* g = src + (t * 4 + k * 1024);
        asm volatile("global_load_async_to_lds_b128 %0, %1, off"
                     :
                     : "v"(ldso), "v"(g)
                     : "memory");
    }
    // Wait for this wave's async transfers to land in LDS, then sync the WG.
    asm volatile("s_wait_asynccnt 0" ::: "memory");
    __syncthreads();

    // ---- Windowed sums: 4 chunks of 4 consecutive outputs per thread ----
    // chunk k: outputs [4t + 1024k, +4). Seed sum (20 adds) + 3 slide steps.
    const float inv = 1.0f / (float)WIN;
#pragma unroll
    for (int k = 0; k < 4; ++k) {
        const int o0 = t * 4 + k * 1024;

        float s0 = 0.0f;
#pragma unroll
        for (int j = 0; j < WIN; ++j) s0 += row[o0 + j];
        float s1 = s0 - row[o0 + 0] + row[o0 + WIN + 0];
        float s2 = s1 - row[o0 + 1] + row[o0 + WIN + 1];
        float s3 = s2 - row[o0 + 2] + row[o0 + WIN + 2];

        // Output rows are only 4B-aligned (4077*4 % 16 != 0) -> scalar stores.
        if (o0 + 0 < TOUT) dst[o0 + 0] = s0 * inv;
        if (o0 + 1 < TOUT) dst[o0 + 1] = s1 * inv;
        if (o0 + 2 < TOUT) dst[o0 + 2] = s2 * inv;
        if (o0 + 3 < TOUT) dst[o0 + 3] = s3 * inv;
    }
}

extern "C" void kernel_launch(void* const* d_in, const int* in_sizes, int n_in,
                              void* d_out, int out_size, void* d_ws, size_t ws_size,
                              hipStream_t stream) {
    (void)n_in; (void)out_size; (void)d_ws; (void)ws_size;
    const float* x = (const float*)d_in[0];
    float* out = (float*)d_out;
    const int nrows = in_sizes[0] / ROW;   // 128*256 = 32768
    TsMean_kernel<<<nrows, NTHREADS, 0, stream>>>(x, out, nrows);
}